// ComplexShiftedWindowAttnBlock_14018773254209
// MI455X (gfx1250) — compile-verified
//
#include <hip/hip_runtime.h>
#include <math.h>
#include <stdint.h>

// ---------------------------------------------------------------------------
// Complex shifted-window attention block for MI455X (gfx1250).
// fp32 complex arithmetic on V_WMMA_F32_16X16X4_F32 (wave32 WMMA).
// Complex GEMM = 4 real WMMA accumulations per k-step per tile:
//   Cr = Ar*Br - Ai*Bi   (two accumulators, VALU subtract in epilogue)
//   Ci = Ar*Bi + Ai*Br   (one accumulator, chained)
// v3: async global->LDS staging fixed to the probed builtin signature
//     (generic v4i* pointers); weight planes deinterleaved once per launch;
//     16x32 tile per wave with register double-buffered fragment loads.
// ---------------------------------------------------------------------------

typedef float v2f __attribute__((ext_vector_type(2)));
typedef float v8f __attribute__((ext_vector_type(8)));
typedef int   v4i __attribute__((ext_vector_type(4)));

#define WMMAF32(A, Bm, Cacc) \
  __builtin_amdgcn_wmma_f32_16x16x4_f32(false, (A), false, (Bm), (short)0, (Cacc), false, false)

#if defined(__AMDGCN__) && __has_builtin(__builtin_amdgcn_global_load_async_to_lds_b128)
#define USE_ASYNC_LDS 1
// probed signature: (v4i* global_src, v4i* lds_dst, imm offset, imm cpol)
#define ASYNC_CP128(gsrc, ldst) \
  __builtin_amdgcn_global_load_async_to_lds_b128((v4i*)(gsrc), (v4i*)(ldst), 0, 0)
#else
#define USE_ASYNC_LDS 0
#endif

__device__ __forceinline__ void wait_async0() {
#if __has_builtin(__builtin_amdgcn_s_wait_asynccnt)
  __builtin_amdgcn_s_wait_asynccnt(0);
#elif defined(__AMDGCN__)
  asm volatile("s_wait_asynccnt 0x0" ::: "memory");
#endif
}

namespace {
constexpr int kB = 4, kH = 56, kW = 56, kC = 256;
constexpr int kHeads = 8, kDh = 32, kWS = 7, kN = 49;
constexpr int kTok = kB * kH * kW;   // 12544 tokens
constexpr int kHid = 4 * kC;         // 1024
}

// ---------------------------------------------------------------------------
// Row (window-major) -> token maps. branch0: plain partition; branch1: shift.
// ---------------------------------------------------------------------------
__global__ void k_rowmap(int* __restrict__ rm0, int* __restrict__ rm1) {
  int r = blockIdx.x * blockDim.x + threadIdx.x;
  if (r >= kTok) return;
  int win = r / kN, n = r % kN;
  int b = win >> 6, wh = (win >> 3) & 7, ww = win & 7;
  int i = n / kWS, j = n % kWS;
  int y = wh * kWS + i, x = ww * kWS + j;
  rm0[r] = (b * kH + y) * kW + x;
  int ys = (y + 3) % kH, xs = (x + 3) % kW;
  rm1[r] = (b * kH + ys) * kW + xs;
}

// Deinterleave complex (re,im) pairs into separate planes.
__global__ void k_deint(const float* __restrict__ src, float* __restrict__ dre,
                        float* __restrict__ dim, int n) {
  int i = blockIdx.x * blockDim.x + threadIdx.x;
  if (i >= n) return;
  float2 v = *(const float2*)(src + 2 * i);
  dre[i] = v.x;
  dim[i] = v.y;
}

// ---------------------------------------------------------------------------
// Complex LayerNorm: LN on re/im separately, complex affine g,b.
// ---------------------------------------------------------------------------
__global__ __launch_bounds__(256)
void k_cln(const float* __restrict__ xre, const float* __restrict__ xim,
           const float* __restrict__ g, const float* __restrict__ bb,
           float* __restrict__ ore, float* __restrict__ oim) {
  int t = blockIdx.x;
  int c = threadIdx.x;
  size_t o = (size_t)t * kC + c;
  float vr = xre[o], vi = xim[o];
  float sr = vr, si = vi, sr2 = vr * vr, si2 = vi * vi;
  for (int d = 16; d > 0; d >>= 1) {
    sr  += __shfl_xor(sr,  d, 32);
    si  += __shfl_xor(si,  d, 32);
    sr2 += __shfl_xor(sr2, d, 32);
    si2 += __shfl_xor(si2, d, 32);
  }
  __shared__ float red[4][8];
  __shared__ float stats[4];
  int lane = threadIdx.x & 31, wid = threadIdx.x >> 5;
  if (lane == 0) { red[0][wid] = sr; red[1][wid] = si; red[2][wid] = sr2; red[3][wid] = si2; }
  __syncthreads();
  if (threadIdx.x == 0) {
    float a = 0, b2 = 0, c2 = 0, d2 = 0;
    for (int w = 0; w < 8; ++w) { a += red[0][w]; b2 += red[1][w]; c2 += red[2][w]; d2 += red[3][w]; }
    stats[0] = a / kC; stats[1] = b2 / kC; stats[2] = c2 / kC; stats[3] = d2 / kC;
  }
  __syncthreads();
  float mur = stats[0], mui = stats[1];
  float varr = stats[2] - mur * mur, vari = stats[3] - mui * mui;
  float lr = (vr - mur) * rsqrtf(varr + 1e-5f);
  float li = (vi - mui) * rsqrtf(vari + 1e-5f);
  float gr = g[2 * c], gi = g[2 * c + 1], br = bb[2 * c], bi = bb[2 * c + 1];
  ore[o] = lr * gr - li * gi + br;
  oim[o] = lr * gi + li * gr + bi;
}

// ---------------------------------------------------------------------------
// Complex GEMM: O[m,n] = sum_k A[m,k]*W[n,k] + bias[n].
// A re/im planes (optional rowmap gather), W re/im planes, bias interleaved.
// One wave computes a 16x32 tile (2 N-tiles share the A fragment).
// Fragment loads are register double-buffered: next k-step's 6 float2 loads
// are issued before the current 8 WMMAs so global latency overlaps compute.
// ---------------------------------------------------------------------------
__global__ __launch_bounds__(256)
void k_cgemm(const float* __restrict__ Are, const float* __restrict__ Aim,
             const float* __restrict__ Wre, const float* __restrict__ Wim,
             const float* __restrict__ Bc,
             const int* __restrict__ rowmap,
             float* __restrict__ Ore, float* __restrict__ Oim,
             int M, int K, int Nn) {
  int wave = (blockIdx.x * blockDim.x + threadIdx.x) >> 5;
  int np = Nn >> 5;                         // number of 32-wide column pairs
  int total = (M >> 4) * np;
  if (wave >= total) return;                // wave-uniform exit
  int tm = wave / np, tp = wave % np;
  int lane = threadIdx.x & 31;
  int koff = (lane >> 4) << 1;              // lanes 16..31 hold K+2,K+3
  int row = tm * 16 + (lane & 15);
  int tok = rowmap ? rowmap[row] : row;
  const float* ar = Are + (size_t)tok * K + koff;
  const float* ai = Aim + (size_t)tok * K + koff;
  int col0 = tp * 32 + (lane & 15);
  int col1 = col0 + 16;
  const float* wr0 = Wre + (size_t)col0 * K + koff;
  const float* wi0 = Wim + (size_t)col0 * K + koff;
  const float* wr1 = Wre + (size_t)col1 * K + koff;
  const float* wi1 = Wim + (size_t)col1 * K + koff;

  v8f aRR0 = {0.f,0.f,0.f,0.f,0.f,0.f,0.f,0.f};
  v8f aII0 = {0.f,0.f,0.f,0.f,0.f,0.f,0.f,0.f};
  v8f aI0  = {0.f,0.f,0.f,0.f,0.f,0.f,0.f,0.f};
  v8f aRR1 = {0.f,0.f,0.f,0.f,0.f,0.f,0.f,0.f};
  v8f aII1 = {0.f,0.f,0.f,0.f,0.f,0.f,0.f,0.f};
  v8f aI1  = {0.f,0.f,0.f,0.f,0.f,0.f,0.f,0.f};

  float2 cA  = *(const float2*)(ar);
  float2 cAi = *(const float2*)(ai);
  float2 cR0 = *(const float2*)(wr0);
  float2 cI0 = *(const float2*)(wi0);
  float2 cR1 = *(const float2*)(wr1);
  float2 cI1 = *(const float2*)(wi1);

#pragma unroll 2
  for (int k = 0; k < K; k += 4) {
    int kn = k + 4;
    int kp = (kn < K) ? kn : 0;            // dummy (valid) address on last iter
    float2 nA  = *(const float2*)(ar + kp);
    float2 nAi = *(const float2*)(ai + kp);
    float2 nR0 = *(const float2*)(wr0 + kp);
    float2 nI0 = *(const float2*)(wi0 + kp);
    float2 nR1 = *(const float2*)(wr1 + kp);
    float2 nI1 = *(const float2*)(wi1 + kp);

    v2f aR  = {cA.x,  cA.y};
    v2f aI  = {cAi.x, cAi.y};
    v2f bR0 = {cR0.x, cR0.y};
    v2f bI0 = {cI0.x, cI0.y};
    v2f bR1 = {cR1.x, cR1.y};
    v2f bI1 = {cI1.x, cI1.y};

    aRR0 = WMMAF32(aR, bR0, aRR0);
    aII0 = WMMAF32(aI, bI0, aII0);
    aI0  = WMMAF32(aR, bI0, aI0);
    aI0  = WMMAF32(aI, bR0, aI0);
    aRR1 = WMMAF32(aR, bR1, aRR1);
    aII1 = WMMAF32(aI, bI1, aII1);
    aI1  = WMMAF32(aR, bI1, aI1);
    aI1  = WMMAF32(aI, bR1, aI1);

    cA = nA; cAi = nAi; cR0 = nR0; cI0 = nI0; cR1 = nR1; cI1 = nI1;
  }

  float br0 = Bc[2 * col0], bi0 = Bc[2 * col0 + 1];
  float br1 = Bc[2 * col1], bi1 = Bc[2 * col1 + 1];
  int mbase = tm * 16 + ((lane >> 4) << 3);
  for (int r = 0; r < 8; ++r) {
    size_t o0 = (size_t)(mbase + r) * Nn + col0;
    Ore[o0]      = aRR0[r] - aII0[r] + br0;
    Oim[o0]      = aI0[r] + bi0;
    Ore[o0 + 16] = aRR1[r] - aII1[r] + br1;
    Oim[o0 + 16] = aI1[r] + bi1;
  }
}

// ---------------------------------------------------------------------------
// Windowed complex attention, one block (4 waves) per (window, head).
// q/k/v staged into LDS via GLOBAL_LOAD_ASYNC_TO_LDS_B128 when available.
// Scale 1/sqrt(d) applied in the S epilogue (async DMA cannot scale in-flight).
// ---------------------------------------------------------------------------
__global__ __launch_bounds__(128)
void k_attn(const float* __restrict__ Qre, const float* __restrict__ Qim,  // [tok,768]
            const float* __restrict__ rpb,                                  // (169*8) cplx
            float* __restrict__ Ore, float* __restrict__ Oim,               // [tok,256]
            int masked) {
  int win = blockIdx.x >> 3;
  int h = blockIdx.x & 7;
  __shared__ float sm[12288];                 // 48 KB
  float* QR = sm;                             // 64x32 each
  float* QI = sm + 2048;
  float* KR = sm + 4096;
  float* KI = sm + 6144;
  float* VR = sm + 8192;
  float* VI = sm + 10240;
  float* SR = sm;                             // 64x64, overlaps QR..KI
  float* SI = sm + 4096;
  const float scale = 0.17677669529663687f;   // 1/sqrt(32)
  int base = win * kN;

#if USE_ASYNC_LDS
  for (int idx = threadIdx.x; idx < 512; idx += 128) {
    int r = idx >> 3, d4 = (idx & 7) << 2;    // 4 floats (16B) per transfer
    int li = r * 32 + d4;
    if (r < kN) {
      size_t o = (size_t)(base + r) * 768 + h * kDh + d4;
      ASYNC_CP128(Qre + o,          &QR[li]);
      ASYNC_CP128(Qim + o,          &QI[li]);
      ASYNC_CP128(Qre + o + kC,     &KR[li]);
      ASYNC_CP128(Qim + o + kC,     &KI[li]);
      ASYNC_CP128(Qre + o + 2 * kC, &VR[li]);
      ASYNC_CP128(Qim + o + 2 * kC, &VI[li]);
    } else {
      float4 z = {0.f, 0.f, 0.f, 0.f};
      *(float4*)&QR[li] = z; *(float4*)&QI[li] = z;
      *(float4*)&KR[li] = z; *(float4*)&KI[li] = z;
      *(float4*)&VR[li] = z; *(float4*)&VI[li] = z;
    }
  }
  wait_async0();
#else
  for (int idx = threadIdx.x; idx < 2048; idx += 128) {
    int r = idx >> 5, d = idx & 31;
    float qr = 0, qi = 0, kr = 0, ki = 0, vr = 0, vi = 0;
    if (r < kN) {
      size_t o = (size_t)(base + r) * 768 + h * kDh + d;
      qr = Qre[o];          qi = Qim[o];
      kr = Qre[o + kC];     ki = Qim[o + kC];
      vr = Qre[o + 2 * kC]; vi = Qim[o + 2 * kC];
    }
    QR[idx] = qr; QI[idx] = qi; KR[idx] = kr; KI[idx] = ki; VR[idx] = vr; VI[idx] = vi;
  }
#endif
  __syncthreads();

  int lane = threadIdx.x & 31;
  int mtile = threadIdx.x >> 5;
  int koff = (lane >> 4) << 1;
  int arow = mtile * 16 + (lane & 15);

  v8f sRR[4] = {}, sII[4] = {}, sI[4] = {};
  for (int t = 0; t < 4; ++t) {
    int bcol = t * 16 + (lane & 15);
    for (int k = 0; k < kDh; k += 4) {
      v2f aR = {QR[arow * 32 + k + koff], QR[arow * 32 + k + koff + 1]};
      v2f aI = {QI[arow * 32 + k + koff], QI[arow * 32 + k + koff + 1]};
      v2f bR = {KR[bcol * 32 + k + koff], KR[bcol * 32 + k + koff + 1]};
      v2f bI = {KI[bcol * 32 + k + koff], KI[bcol * 32 + k + koff + 1]};
      sRR[t] = WMMAF32(aR, bR, sRR[t]);
      sII[t] = WMMAF32(aI, bI, sII[t]);
      sI[t]  = WMMAF32(aR, bI, sI[t]);
      sI[t]  = WMMAF32(aI, bR, sI[t]);
    }
  }
  __syncthreads();   // q/k reads done; S may overwrite that LDS region

  int wh = (win >> 3) & 7, ww = win & 7;
  for (int t = 0; t < 4; ++t) {
    int m = t * 16 + (lane & 15);
    for (int r = 0; r < 8; ++r) {
      int n = mtile * 16 + ((lane >> 4) << 3) + r;
      float sre, sim;
      if (n < kN && m < kN) {
        int i1 = n / kWS, j1 = n % kWS, i2 = m / kWS, j2 = m % kWS;
        int ridx = (i1 - i2 + 6) * 13 + (j1 - j2 + 6);
        float rr = rpb[(ridx * kHeads + h) * 2];
        float ri = rpb[(ridx * kHeads + h) * 2 + 1];
        sre = (sRR[t][r] - sII[t][r]) * scale + rr;
        sim = sI[t][r] * scale + ri;
        if (masked) {
          int y1 = wh * kWS + i1, x1 = ww * kWS + j1;
          int y2 = wh * kWS + i2, x2 = ww * kWS + j2;
          int l1 = (y1 < 49 ? 0 : (y1 < 53 ? 1 : 2)) * 3 + (x1 < 49 ? 0 : (x1 < 53 ? 1 : 2));
          int l2 = (y2 < 49 ? 0 : (y2 < 53 ? 1 : 2)) * 3 + (x2 < 49 ? 0 : (x2 < 53 ? 1 : 2));
          if (l1 != l2) sre -= 100.f;   // mask adds to real part only
        }
      } else {
        sre = -1e30f; sim = -1e30f;     // padded cols vanish under softmax
      }
      SR[n * 64 + m] = sre;
      SI[n * 64 + m] = sim;
    }
  }
  __syncthreads();

  if (threadIdx.x < 64) {               // row softmax, re and im independently
    int n = threadIdx.x;
    for (int p = 0; p < 2; ++p) {
      float* S = p ? SI : SR;
      float mx = -3e38f;
      for (int m = 0; m < 64; ++m) mx = fmaxf(mx, S[n * 64 + m]);
      float sum = 0.f;
      for (int m = 0; m < 64; ++m) { float e = __expf(S[n * 64 + m] - mx); S[n * 64 + m] = e; sum += e; }
      float inv = 1.f / sum;
      for (int m = 0; m < 64; ++m) S[n * 64 + m] *= inv;
    }
  }
  __syncthreads();

  for (int t = 0; t < 2; ++t) {         // out = S @ V   (d = 32 -> 2 N-tiles)
    v8f oRR = {0.f,0.f,0.f,0.f,0.f,0.f,0.f,0.f};
    v8f oII = {0.f,0.f,0.f,0.f,0.f,0.f,0.f,0.f};
    v8f oI  = {0.f,0.f,0.f,0.f,0.f,0.f,0.f,0.f};
    int bcol = t * 16 + (lane & 15);
    for (int k = 0; k < 64; k += 4) {
      v2f aR = {SR[arow * 64 + k + koff], SR[arow * 64 + k + koff + 1]};
      v2f aI = {SI[arow * 64 + k + koff], SI[arow * 64 + k + koff + 1]};
      v2f bR = {VR[(k + koff) * 32 + bcol], VR[(k + koff + 1) * 32 + bcol]};
      v2f bI = {VI[(k + koff) * 32 + bcol], VI[(k + koff + 1) * 32 + bcol]};
      oRR = WMMAF32(aR, bR, oRR);
      oII = WMMAF32(aI, bI, oII);
      oI  = WMMAF32(aR, bI, oI);
      oI  = WMMAF32(aI, bR, oI);
    }
    for (int r = 0; r < 8; ++r) {
      int n = mtile * 16 + ((lane >> 4) << 3) + r;
      if (n < kN) {
        size_t o = (size_t)(base + n) * kC + h * kDh + bcol;
        Ore[o] = oRR[r] - oII[r];
        Oim[o] = oI[r];
      }
    }
  }
}

// ---------------------------------------------------------------------------
// Elementwise helpers
// ---------------------------------------------------------------------------
__global__ void k_cgelu(float* __restrict__ re, float* __restrict__ im, size_t n) {
  size_t i = (size_t)blockIdx.x * blockDim.x + threadIdx.x;
  if (i >= n) return;
  float xr = re[i], xi = im[i];
  float tr = 0.7978845608028654f * (xr + 0.044715f * xr * xr * xr);
  float ti = 0.7978845608028654f * (xi + 0.044715f * xi * xi * xi);
  re[i] = 0.5f * xr * (1.f + tanhf(tr));
  im[i] = 0.5f * xi * (1.f + tanhf(ti));
}

// out[tok] = base[tok] + add[window-row], scattered through rowmap (bijection)
__global__ void k_scatter_add(const float* __restrict__ bre, const float* __restrict__ bim,
                              const float* __restrict__ are, const float* __restrict__ aim,
                              const int* __restrict__ rmap,
                              float* __restrict__ ore, float* __restrict__ oim) {
  int idx = blockIdx.x * blockDim.x + threadIdx.x;
  if (idx >= kTok * kC) return;
  int r = idx >> 8, c = idx & 255;
  size_t to = (size_t)rmap[r] * kC + c;
  ore[to] = bre[to] + are[idx];
  oim[to] = bim[to] + aim[idx];
}

__global__ void k_add2(const float* __restrict__ are, const float* __restrict__ aim,
                       const float* __restrict__ bre, const float* __restrict__ bim,
                       float* __restrict__ ore, float* __restrict__ oim) {
  int i = blockIdx.x * blockDim.x + threadIdx.x;
  if (i >= kTok * kC) return;
  ore[i] = are[i] + bre[i];
  oim[i] = aim[i] + bim[i];
}

__global__ void k_final(const float* __restrict__ xre, const float* __restrict__ xim,
                        const float* __restrict__ mre, const float* __restrict__ mim,
                        float* __restrict__ out) {
  int i = blockIdx.x * blockDim.x + threadIdx.x;
  if (i >= kTok * kC) return;
  out[2 * i]     = xre[i] + mre[i];
  out[2 * i + 1] = xim[i] + mim[i];
}

// ---------------------------------------------------------------------------
// Host-side orchestration
// ---------------------------------------------------------------------------
extern "C" void kernel_launch(void* const* d_in, const int* in_sizes, int n_in,
                              void* d_out, int out_size, void* d_ws, size_t ws_size,
                              hipStream_t stream) {
  (void)in_sizes; (void)n_in; (void)out_size; (void)ws_size;
  // d_in order: x_re, x_im, then params flattened in JAX pytree (sorted-key) order.
  const float* x_re      = (const float*)d_in[0];
  const float* x_im      = (const float*)d_in[1];
  const float* sw_proj_b = (const float*)d_in[2];
  const float* sw_proj_w = (const float*)d_in[3];
  const float* sw_qkv_b  = (const float*)d_in[4];
  const float* sw_qkv_w  = (const float*)d_in[5];
  const float* sw_rpb    = (const float*)d_in[6];
  const float* sw_fc1_b  = (const float*)d_in[7];
  const float* sw_fc1_w  = (const float*)d_in[8];
  const float* sw_fc2_b  = (const float*)d_in[9];
  const float* sw_fc2_w  = (const float*)d_in[10];
  const float* sw_n1_b   = (const float*)d_in[11];
  const float* sw_n1_g   = (const float*)d_in[12];
  const float* sw_n2_b   = (const float*)d_in[13];
  const float* sw_n2_g   = (const float*)d_in[14];
  const float* w_proj_b  = (const float*)d_in[15];
  const float* w_proj_w  = (const float*)d_in[16];
  const float* w_qkv_b   = (const float*)d_in[17];
  const float* w_qkv_w   = (const float*)d_in[18];
  const float* w_rpb     = (const float*)d_in[19];
  const float* w_fc1_b   = (const float*)d_in[20];
  const float* w_fc1_w   = (const float*)d_in[21];
  const float* w_fc2_b   = (const float*)d_in[22];
  const float* w_fc2_w   = (const float*)d_in[23];
  const float* w_n1_b    = (const float*)d_in[24];
  const float* w_n1_g    = (const float*)d_in[25];
  const float* w_n2_b    = (const float*)d_in[26];
  const float* w_n2_g    = (const float*)d_in[27];

  const size_t TC = (size_t)kTok * kC;        // 3,211,264
  const size_t TH = (size_t)kTok * kHid;      // 12,845,056
  constexpr int NQKV = 3 * kC * kC;           // 196608
  constexpr int NPJ  = kC * kC;               // 65536
  constexpr int NF1  = kHid * kC;             // 262144
  constexpr int NF2  = kC * kHid;             // 262144

  float* p = (float*)d_ws;
  float* xn_re  = p;             float* xn_im  = xn_re + TC;
  float* big_re = xn_im + TC;    float* big_im = big_re + TH;   // qkv(768) or h1(1024)
  float* ao_re  = big_im + TH;   float* ao_im  = ao_re + TC;
  float* pr_re  = ao_im + TC;    float* pr_im  = pr_re + TC;
  float* x1_re  = pr_im + TC;    float* x1_im  = x1_re + TC;
  float* x2_re  = x1_im + TC;    float* x2_im  = x2_re + TC;
  // deinterleaved weight planes
  float* wqv_re  = x2_im + TC;      float* wqv_im  = wqv_re + NQKV;
  float* wpj_re  = wqv_im + NQKV;   float* wpj_im  = wpj_re + NPJ;
  float* wf1_re  = wpj_im + NPJ;    float* wf1_im  = wf1_re + NF1;
  float* wf2_re  = wf1_im + NF1;    float* wf2_im  = wf2_re + NF2;
  float* sqv_re  = wf2_im + NF2;    float* sqv_im  = sqv_re + NQKV;
  float* spj_re  = sqv_im + NQKV;   float* spj_im  = spj_re + NPJ;
  float* sf1_re  = spj_im + NPJ;    float* sf1_im  = sf1_re + NF1;
  float* sf2_re  = sf1_im + NF1;    float* sf2_im  = sf2_re + NF2;
  int* rm0 = (int*)(sf2_im + NF2);
  int* rm1 = rm0 + kTok;

  const int EW = (kTok * kC) / 256;           // 12544 blocks for [TOK,C] elementwise

  auto deint = [&](const float* src, float* dre, float* dim, int n) {
    k_deint<<<(n + 255) / 256, 256, 0, stream>>>(src, dre, dim, n);
  };
  auto gemm = [&](const float* Ar, const float* Ai, const float* Wr, const float* Wi,
                  const float* bias, const int* rmap, float* Or, float* Oi, int K, int Nn) {
    int waves = (kTok / 16) * (Nn / 32);      // always a multiple of 8
    k_cgemm<<<waves / 8, 256, 0, stream>>>(Ar, Ai, Wr, Wi, bias, rmap, Or, Oi, kTok, K, Nn);
  };

  k_rowmap<<<(kTok + 255) / 256, 256, 0, stream>>>(rm0, rm1);
  deint(w_qkv_w,  wqv_re, wqv_im, NQKV);
  deint(w_proj_w, wpj_re, wpj_im, NPJ);
  deint(w_fc1_w,  wf1_re, wf1_im, NF1);
  deint(w_fc2_w,  wf2_re, wf2_im, NF2);
  deint(sw_qkv_w,  sqv_re, sqv_im, NQKV);
  deint(sw_proj_w, spj_re, spj_im, NPJ);
  deint(sw_fc1_w,  sf1_re, sf1_im, NF1);
  deint(sw_fc2_w,  sf2_re, sf2_im, NF2);

  // ---- branch 1: regular window attention ----
  k_cln<<<kTok, 256, 0, stream>>>(x_re, x_im, w_n1_g, w_n1_b, xn_re, xn_im);
  gemm(xn_re, xn_im, wqv_re, wqv_im, w_qkv_b, rm0, big_re, big_im, kC, 3 * kC);
  k_attn<<<256 * kHeads, 128, 0, stream>>>(big_re, big_im, w_rpb, ao_re, ao_im, 0);
  gemm(ao_re, ao_im, wpj_re, wpj_im, w_proj_b, nullptr, pr_re, pr_im, kC, kC);
  k_scatter_add<<<EW, 256, 0, stream>>>(x_re, x_im, pr_re, pr_im, rm0, x1_re, x1_im);
  // branch-1 MLP
  k_cln<<<kTok, 256, 0, stream>>>(x1_re, x1_im, w_n2_g, w_n2_b, xn_re, xn_im);
  gemm(xn_re, xn_im, wf1_re, wf1_im, w_fc1_b, nullptr, big_re, big_im, kC, kHid);
  k_cgelu<<<(int)((TH + 255) / 256), 256, 0, stream>>>(big_re, big_im, TH);
  gemm(big_re, big_im, wf2_re, wf2_im, w_fc2_b, nullptr, pr_re, pr_im, kHid, kC);
  k_add2<<<EW, 256, 0, stream>>>(x1_re, x1_im, pr_re, pr_im, x2_re, x2_im);

  // ---- branch 2: shifted window attention ----
  k_cln<<<kTok, 256, 0, stream>>>(x2_re, x2_im, sw_n1_g, sw_n1_b, xn_re, xn_im);
  gemm(xn_re, xn_im, sqv_re, sqv_im, sw_qkv_b, rm1, big_re, big_im, kC, 3 * kC);
  k_attn<<<256 * kHeads, 128, 0, stream>>>(big_re, big_im, sw_rpb, ao_re, ao_im, 1);
  gemm(ao_re, ao_im, spj_re, spj_im, sw_proj_b, nullptr, pr_re, pr_im, kC, kC);
  // NOTE: reference residual uses the *normed* x (xn), not x2
  k_scatter_add<<<EW, 256, 0, stream>>>(xn_re, xn_im, pr_re, pr_im, rm1, x1_re, x1_im);
  // branch-2 MLP + final interleaved complex output
  k_cln<<<kTok, 256, 0, stream>>>(x1_re, x1_im, sw_n2_g, sw_n2_b, xn_re, xn_im);
  gemm(xn_re, xn_im, sf1_re, sf1_im, sw_fc1_b, nullptr, big_re, big_im, kC, kHid);
  k_cgelu<<<(int)((TH + 255) / 256), 256, 0, stream>>>(big_re, big_im, TH);
  gemm(big_re, big_im, sf2_re, sf2_im, sw_fc2_b, nullptr, pr_re, pr_im, kHid, kC);
  k_final<<<EW, 256, 0, stream>>>(x1_re, x1_im, pr_re, pr_im, (float*)d_out);
}